// GCNLayer_80633716015134
// MI455X (gfx1250) — compile-verified
//
#include <hip/hip_runtime.h>
#include <hip/hip_bf16.h>

// ---------------------------------------------------------------------------
// GCN layer for MI455X (gfx1250, wave32).
//
// Cost model: GEMM = 100000x32x32 (~205 MFLOP, negligible). Edge aggregation
// dominates: 1.6M edges * 128B gather + 51.2M f32 atomics, all hitting the
// 192MB L2 (h/agg are 12.8MB each). Projection uses V_WMMA_F32_16X16X4_F32
// (full fp32 precision, no downcast risk).
// ---------------------------------------------------------------------------

typedef float v2f __attribute__((ext_vector_type(2)));
typedef float v8f __attribute__((ext_vector_type(8)));

#define IN_FEATS  32
#define OUT_FEATS 32

// ---- Kernel 1: zero output accumulator, init degree to 1.0 (self-loop) ----
__global__ void gcn_init(float* __restrict__ out, float* __restrict__ deg,
                         int n_nodes) {
  int i = blockIdx.x * blockDim.x + threadIdx.x;
  if (i < n_nodes * OUT_FEATS) out[i] = 0.0f;
  if (i < n_nodes) deg[i] = 1.0f;  // self-loop contribution to in-degree
}

// ---- Kernel 2: in-degree count over dst edges -----------------------------
__global__ void gcn_degree(const int* __restrict__ dst,
                           float* __restrict__ deg, int n_edges) {
  int e = blockIdx.x * blockDim.x + threadIdx.x;
  if (e < n_edges) atomicAdd(&deg[dst[e]], 1.0f);
}

// ---- Kernel 3: deg -> 1/sqrt(max(deg,1)) in place -------------------------
__global__ void gcn_rsqrt(float* __restrict__ deg, int n_nodes) {
  int i = blockIdx.x * blockDim.x + threadIdx.x;
  if (i < n_nodes) deg[i] = rsqrtf(fmaxf(deg[i], 1.0f));
}

// ---- Kernel 4: h = (x * inv_sqrt_deg) @ W^T via V_WMMA_F32_16X16X4_F32 ----
// One wave computes a 16-node x 16-outfeat tile; K=32 -> 8 WMMA steps of K=4;
// two N-tiles cover OUT_FEATS=32. Fragment layouts per CDNA5 ISA 7.12.2:
//   A (16x4 f32, 2 VGPRs/lane): M = lane&15, K = k0 + 2*(lane>>4) + {0,1}
//   B (4x16 f32, 2 VGPRs/lane): N = lane&15, K = k0 + 2*(lane>>4) + {0,1}
//   C/D (16x16 f32, 8 VGPRs):   M = r + 8*(lane>>4), N = lane&15
__global__ void gcn_gemm_wmma(const float* __restrict__ feat,
                              const float* __restrict__ W,      // [OUT][IN]
                              const float* __restrict__ inv_sq, // [N]
                              float* __restrict__ h,            // [N][OUT]
                              int n_nodes) {
  const int lane = threadIdx.x & 31;
  const int wave = threadIdx.x >> 5;
  const int tile = blockIdx.x * (blockDim.x >> 5) + wave;
  const int m0 = tile * 16;
  if (m0 >= n_nodes) return;  // whole-wave exit: EXEC stays all-ones for WMMA

  const int row  = lane & 15;         // M index (A) / N index (B)
  const int kb   = (lane >> 4) << 1;  // K sub-offset: 0 (lanes 0-15) or 2
  const int node = m0 + row;
  const float inv = inv_sq[node];

  // Preload all A fragments (row of x, scaled by inv_sqrt(deg)).
  v2f afrag[8];
#pragma unroll
  for (int kk = 0; kk < 8; ++kk) {
    const float* ap = feat + node * IN_FEATS + kk * 4 + kb;
    afrag[kk].x = ap[0] * inv;
    afrag[kk].y = ap[1] * inv;
  }

#pragma unroll
  for (int nt = 0; nt < 2; ++nt) {
    const int col = nt * 16 + row;  // output feature index for B fragment
    v8f c = {0.f, 0.f, 0.f, 0.f, 0.f, 0.f, 0.f, 0.f};
#pragma unroll
    for (int kk = 0; kk < 8; ++kk) {
      // B[k][n] = W^T[k][n] = W[n][k]
      const float* bp = W + col * IN_FEATS + kk * 4 + kb;
      v2f b;
      b.x = bp[0];
      b.y = bp[1];
      c = __builtin_amdgcn_wmma_f32_16x16x4_f32(
          /*neg_a=*/false, afrag[kk], /*neg_b=*/false, b,
          /*c_mod=*/(short)0, c, /*reuse_a=*/false, /*reuse_b=*/false);
    }
    const int mbase = m0 + ((lane >> 4) << 3);
    const int ncol  = nt * 16 + row;
#pragma unroll
    for (int r = 0; r < 8; ++r) {
      h[(mbase + r) * OUT_FEATS + ncol] = c[r];
    }
  }
}

// ---- Kernel 5: edge scatter agg[dst] += h[src] ----------------------------
// 8 threads per edge, one float4 (4 feats) each: coalesced 128B gathers of
// h[src] and 4 global_atomic_add_f32 per thread (L2-resident accumulators).
__global__ void gcn_scatter(const int* __restrict__ src,
                            const int* __restrict__ dst,
                            const float* __restrict__ h,
                            float* __restrict__ agg, int n_edges) {
  int idx = blockIdx.x * blockDim.x + threadIdx.x;
  if (idx >= n_edges * 8) return;
  const int e = idx >> 3;
  const int q = idx & 7;
  const int s = src[e];
  const int d = dst[e];
  const float4 v = *reinterpret_cast<const float4*>(h + s * OUT_FEATS + q * 4);
  float* o = agg + d * OUT_FEATS + q * 4;
  atomicAdd(o + 0, v.x);
  atomicAdd(o + 1, v.y);
  atomicAdd(o + 2, v.z);
  atomicAdd(o + 3, v.w);
}

// ---- Kernel 6: out = relu((agg + h_self) * inv_sqrt_deg + bias) -----------
__global__ void gcn_final(float* __restrict__ out, const float* __restrict__ h,
                          const float* __restrict__ inv_sq,
                          const float* __restrict__ bias, int n_nodes) {
  int i = blockIdx.x * blockDim.x + threadIdx.x;
  if (i >= n_nodes * OUT_FEATS) return;
  const int n = i >> 5;
  const int f = i & 31;
  const float v = (out[i] + h[i]) * inv_sq[n] + bias[f];
  out[i] = fmaxf(v, 0.0f);
}

// ---------------------------------------------------------------------------
extern "C" void kernel_launch(void* const* d_in, const int* in_sizes, int n_in,
                              void* d_out, int out_size, void* d_ws,
                              size_t ws_size, hipStream_t stream) {
  (void)n_in; (void)out_size; (void)ws_size;
  const float* feat = (const float*)d_in[0];  // [N, 32]
  const int*   src  = (const int*)d_in[1];    // [E]
  const int*   dst  = (const int*)d_in[2];    // [E]
  const float* W    = (const float*)d_in[3];  // [32, 32]
  const float* bias = (const float*)d_in[4];  // [32]
  float* out = (float*)d_out;                 // [N, 32], used as agg accum

  const int n_nodes = in_sizes[0] / IN_FEATS;
  const int n_edges = in_sizes[1];

  // Workspace: deg/inv_sqrt [N] floats, then h [N,32] floats (16B aligned:
  // N*4 = 400000 bytes is a multiple of 16).
  float* deg = (float*)d_ws;
  float* h   = deg + n_nodes;

  const int B = 256;
  const int nf = n_nodes * OUT_FEATS;

  gcn_init<<<(nf + B - 1) / B, B, 0, stream>>>(out, deg, n_nodes);
  gcn_degree<<<(n_edges + B - 1) / B, B, 0, stream>>>(dst, deg, n_edges);
  gcn_rsqrt<<<(n_nodes + B - 1) / B, B, 0, stream>>>(deg, n_nodes);

  // 8 waves per block -> 8 node-tiles of 16 nodes per block.
  const int tiles = (n_nodes + 15) / 16;
  gcn_gemm_wmma<<<(tiles + 7) / 8, B, 0, stream>>>(feat, W, deg, h, n_nodes);

  gcn_scatter<<<(n_edges * 8 + B - 1) / B, B, 0, stream>>>(src, dst, h, out,
                                                           n_edges);
  gcn_final<<<(nf + B - 1) / B, B, 0, stream>>>(out, h, deg, bias, n_nodes);
}